// CombinedLoss_6493990552086
// MI455X (gfx1250) — compile-verified
//
#include <hip/hip_runtime.h>
#include <hip/hip_bf16.h>
#include <math.h>

// ---------------------------------------------------------------------------
// CombinedLoss for MI455X (gfx1250, wave32).
// Dominant cost: (12000x512) @ (512x4096) f32 GEMM (50.3 GFLOP) fused with
// online logsumexp (CE), argmax-excluding-target (hard negative mining), and
// target-logit capture, via V_WMMA_F32_16X16X4_F32. Codebook (8MB) is
// L2-resident; the 12000x4096 logit matrix is never materialized.
// Epilogue uses integer-mask arithmetic (cannot be unfolded into branches).
// All logit-side state is carried pre-scaled by 1/CE_TEMP.
// ---------------------------------------------------------------------------

typedef __attribute__((ext_vector_type(2))) float v2f;
typedef __attribute__((ext_vector_type(8))) float v8f;

#define NB 8
#define CC 512
#define TT 1500
#define KK 4096
#define NN (NB * TT)          // 12000 rows, = 750 * 16
#define INVT 10.0f            // 1 / CE_TEMP
#define NBLK_ROWLOSS 47       // ceil(12000/256)

__device__ __forceinline__ float fbits_or(float a, unsigned msk) {
  return __uint_as_float(__float_as_uint(a) | msk);
}

// ---------------- prep: c2[k] = ||codebook[k]||^2 (wave per k) -------------
__global__ __launch_bounds__(256) void c2_kernel(const float* __restrict__ cb,
                                                 float* __restrict__ c2) {
  int w = (blockIdx.x * blockDim.x + threadIdx.x) >> 5;   // 4096 waves
  int lane = threadIdx.x & 31;
  const float* row = cb + (size_t)w * CC;
  float sum = 0.f;
  for (int c = lane; c < CC; c += 32) { float v = row[c]; sum += v * v; }
  #pragma unroll
  for (int o = 16; o; o >>= 1) sum += __shfl_xor(sum, o, 32);
  if (lane == 0) c2[w] = sum;
}

// -------- prep: cbT2[(c/2)*4096 + k] = float2{cb[k][c], cb[k][c+1]} --------
__global__ __launch_bounds__(256) void cbt_kernel(const float* __restrict__ cb,
                                                  float* __restrict__ cbT2) {
  __shared__ float tile[64][65];
  int kb = blockIdx.x * 64, c0 = blockIdx.y * 64;
  int tx = threadIdx.x & 63, ty = threadIdx.x >> 6;       // 64 x 4
  #pragma unroll
  for (int i = 0; i < 16; i++) {
    int kl = i * 4 + ty;
    tile[kl][tx] = cb[(size_t)(kb + kl) * CC + c0 + tx];  // coalesced along c
  }
  __syncthreads();
  float2* dst = (float2*)cbT2;
  #pragma unroll
  for (int i = 0; i < 8; i++) {
    int c2l = i * 4 + ty;                                  // 0..31 (c-pair)
    int kl = tx;                                           // 0..63
    float2 v = make_float2(tile[kl][2 * c2l], tile[kl][2 * c2l + 1]);
    dst[(size_t)((c0 >> 1) + c2l) * KK + kb + kl] = v;     // coalesced along k
  }
}

// -------- prep: zf[n*512+c] = student[b, c, t]  (n = b*1500 + t) -----------
__global__ __launch_bounds__(256) void zt_kernel(const float* __restrict__ stu,
                                                 float* __restrict__ zf) {
  __shared__ float tile[64][65];
  int tb = blockIdx.x * 64, c0 = blockIdx.y * 64, b = blockIdx.z;
  int tx = threadIdx.x & 63, ty = threadIdx.x >> 6;
  const size_t CT = (size_t)CC * TT;
  #pragma unroll
  for (int i = 0; i < 16; i++) {
    int cl = i * 4 + ty;
    int t = tb + tx;
    tile[cl][tx] = (t < TT) ? stu[(size_t)b * CT + (size_t)(c0 + cl) * TT + t] : 0.f;
  }
  __syncthreads();
  #pragma unroll
  for (int i = 0; i < 16; i++) {
    int tl = i * 4 + ty;
    int t = tb + tl;
    if (t < TT) zf[(size_t)(b * TT + t) * CC + c0 + tx] = tile[tx][tl];
  }
}

// ---------------- fused WMMA GEMM + softmax + hard-negative ----------------
// Block = 16 rows x 256 threads (8 waves). Wave w sweeps column tiles
// kt = w, w+8, ... (32 tiles). Per tile: 128 x V_WMMA_F32_16X16X4_F32 over
// C=512, then a mask-arithmetic (branch-proof) epilogue doing online
// logsumexp / argmax(!=tgt) / target-logit capture, all in logit units.
__global__ __launch_bounds__(256) void gemm_kernel(
    const float* __restrict__ zf, const float* __restrict__ cbT2,
    const float* __restrict__ c2, const int* __restrict__ codes,
    float* __restrict__ ce_out, int* __restrict__ hidx_out) {
  __shared__ float ldsA[16 * 516];                 // 16 rows, stride 516 (no bank conflicts)
  __shared__ float ldsM[16][8], ldsS[16][8], ldsB[16][8], ldsQT[16][8];
  __shared__ int   ldsBI[16][8];
  __shared__ int   ldsTG[16];

  const int tid = threadIdx.x;
  const int n0 = blockIdx.x * 16;

  { // stage A tile (fully coalesced float4 from zf)
    int r = tid >> 4, cq = tid & 15;
    const float4* src = (const float4*)(zf + (size_t)(n0 + r) * CC);
    float* drow = ldsA + r * 516;
    #pragma unroll
    for (int i = 0; i < 8; i++) {
      float4 v = src[cq + 16 * i];
      int c = (cq + 16 * i) * 4;
      drow[c] = v.x; drow[c + 1] = v.y; drow[c + 2] = v.z; drow[c + 3] = v.w;
    }
    if (tid < 16) ldsTG[tid] = codes[n0 + tid];
  }
  __syncthreads();

  const int lane = tid & 31, w = tid >> 5;
  const int col = lane & 15, hi = lane >> 4;       // D: col = lane%16, rows j+8*hi
  int tg[8];
  #pragma unroll
  for (int j = 0; j < 8; j++) tg[j] = ldsTG[j + 8 * hi];

  float m[8], s[8], best[8], qt[8];
  int bidx[8];
  #pragma unroll
  for (int j = 0; j < 8; j++) {
    m[j] = -INFINITY; s[j] = 0.f; best[j] = -INFINITY; qt[j] = 0.f; bidx[j] = 0;
  }

  const float* arow = ldsA + col * 516 + 2 * hi;   // A: M=lane%16, K pair {0,1}|{2,3}

  for (int it = 0; it < 32; ++it) {
    const int k0 = (w + 8 * it) * 16;
    v8f acc = {0.f, 0.f, 0.f, 0.f, 0.f, 0.f, 0.f, 0.f};
    const v2f* bptr = ((const v2f*)cbT2) + (size_t)hi * KK + k0 + col;
    #pragma unroll 8
    for (int c0 = 0; c0 < CC; c0 += 4) {
      v2f a = *(const v2f*)(arow + c0);            // ds_load_b64, immediate offsets
      v2f b = bptr[(size_t)(c0 >> 1) * KK];        // global b64, immediate offsets
      acc = __builtin_amdgcn_wmma_f32_16x16x4_f32(
          false, a, false, b, (short)0, acc, false, false);
    }
    const int k = k0 + col;
    const float c2k10 = c2[k] * INVT;
    #pragma unroll
    for (int j = 0; j < 8; j++) {                  // mask-arithmetic epilogue
      float q = fmaf(2.0f * INVT, acc[j], -c2k10); // logit (already / CE_TEMP)
      unsigned tm = (k == tg[j]) ? 0xFFFFFFFFu : 0u;          // v_cndmask
      qt[j] = __uint_as_float(__float_as_uint(qt[j]) |
                              (tm & __float_as_uint(q)));     // capture once, exact
      float qx = __uint_as_float((__float_as_uint(q) & ~tm) |
                                 (tm & 0xFF800000u));         // tgt -> -inf
      bidx[j] = (qx > best[j]) ? k : bidx[j];                 // v_cmp + v_cndmask
      best[j] = fmaxf(best[j], qx);                           // v_max
      float nm = fmaxf(m[j], q);
      s[j] = fmaf(s[j], __expf(m[j] - nm), __expf(q - nm));
      m[j] = nm;
    }
  }

  // reduce across the 16 column-lanes of each half-wave
  #pragma unroll
  for (int off = 1; off < 16; off <<= 1) {
    #pragma unroll
    for (int j = 0; j < 8; j++) {
      float om = __shfl_xor(m[j], off, 32);
      float os = __shfl_xor(s[j], off, 32);
      float ob = __shfl_xor(best[j], off, 32);
      int obi = __shfl_xor(bidx[j], off, 32);
      qt[j] += __shfl_xor(qt[j], off, 32);         // all-zero except owner: exact
      float nm = fmaxf(m[j], om);
      s[j] = fmaf(s[j], __expf(m[j] - nm), os * __expf(om - nm));
      m[j] = nm;
      bool tk = ob > best[j] || (ob == best[j] && obi < bidx[j]);
      best[j] = tk ? ob : best[j];
      bidx[j] = tk ? obi : bidx[j];
    }
  }
  if (col == 0) {
    #pragma unroll
    for (int j = 0; j < 8; j++) {
      int row = j + 8 * hi;
      ldsM[row][w] = m[j]; ldsS[row][w] = s[j];
      ldsB[row][w] = best[j]; ldsBI[row][w] = bidx[j];
      ldsQT[row][w] = qt[j];
    }
  }
  __syncthreads();

  if (tid < 16) {                                  // deterministic cross-wave merge
    float fm = -INFINITY, fs = 0.f, fb = -INFINITY, fqt = 0.f; int fbi = 0;
    for (int wv = 0; wv < 8; wv++) {
      float om = ldsM[tid][wv], os = ldsS[tid][wv];
      float nm = fmaxf(fm, om);
      fs = fmaf(fs, __expf(fm - nm), os * __expf(om - nm));
      fm = nm;
      float ob = ldsB[tid][wv]; int obi = ldsBI[tid][wv];
      bool tk = ob > fb || (ob == fb && obi < fbi);
      fb = tk ? ob : fb;
      fbi = tk ? obi : fbi;
      fqt += ldsQT[tid][wv];
    }
    float lse = fm + __logf(fs);                   // logsumexp of logits
    ce_out[n0 + tid] = lse - fqt;                  // -log p(tgt)
    hidx_out[n0 + tid] = fbi;
  }
}

// --------- streaming per-row losses + block partial reductions -------------
__global__ __launch_bounds__(256) void rowloss_kernel(
    const float* __restrict__ stu, const float* __restrict__ tea,
    const float* __restrict__ org, const float* __restrict__ cbk,
    const float* __restrict__ ce, const int* __restrict__ hidx,
    float* __restrict__ partials) {
  int n = blockIdx.x * 256 + threadIdx.x;
  float dpos2 = 0.f, trip = 0.f, dcos = 0.f, vcnt = 0.f, cev = 0.f;
  if (n < NN) {
    int b = n / TT, t = n % TT;                    // lanes walk t -> coalesced
    const size_t CT = (size_t)CC * TT;
    const float* ps = stu + (size_t)b * CT + t;
    const float* pt = tea + (size_t)b * CT + t;
    const float* po = org + (size_t)b * CT + t;
    const float* pc = cbk + (size_t)hidx[n] * CC;
    float dneg2 = 0.f, mm = 0.f, dd = 0.f, md = 0.f;
    for (int c = 0; c < CC; c++) {
      float sv = ps[(size_t)c * TT], tv = pt[(size_t)c * TT], ov = po[(size_t)c * TT];
      float cv = pc[c];
      float e = tv - sv;  dpos2 += e * e;
      float en = tv - cv; dneg2 += en * en;
      float mv = sv - ov, dv = tv - ov;
      mm += mv * mv; dd += dv * dv; md += mv * dv;
    }
    float dpos = sqrtf(dpos2), dneg = sqrtf(dneg2);
    trip = fmaxf(0.f, dpos - dneg + 0.5f);
    float mn = sqrtf(mm), dn = sqrtf(dd);
    bool valid = (mn > 1e-6f) && (dn > 1e-6f);
    float cosv = md / ((mn + 1e-8f) * (dn + 1e-8f));
    dcos = valid ? (1.f - cosv) : 0.f;
    vcnt = valid ? 1.f : 0.f;
    cev = ce[n];
  }
  __shared__ float red[5][256];
  red[0][threadIdx.x] = dpos2; red[1][threadIdx.x] = trip; red[2][threadIdx.x] = dcos;
  red[3][threadIdx.x] = vcnt;  red[4][threadIdx.x] = cev;
  __syncthreads();
  for (int st = 128; st; st >>= 1) {
    if (threadIdx.x < st)
      #pragma unroll
      for (int q = 0; q < 5; q++) red[q][threadIdx.x] += red[q][threadIdx.x + st];
    __syncthreads();
  }
  if (threadIdx.x < 5) partials[blockIdx.x * 8 + threadIdx.x] = red[threadIdx.x][0];
}

// ------------------------------ finalizer ----------------------------------
__global__ void final_kernel(const float* __restrict__ partials, float* __restrict__ out) {
  if (threadIdx.x == 0) {
    float a[5] = {0.f, 0.f, 0.f, 0.f, 0.f};
    for (int i = 0; i < NBLK_ROWLOSS; i++)
      for (int q = 0; q < 5; q++) a[q] += partials[i * 8 + q];
    float mse = a[0] / ((float)NB * CC * TT);      // feature MSE == dir MSE
    out[0] = 2.f * mse + a[1] / (float)NN + a[4] / (float)NN + a[2] / fmaxf(a[3], 1.f);
  }
}

// ---------------------------------------------------------------------------
extern "C" void kernel_launch(void* const* d_in, const int* in_sizes, int n_in,
                              void* d_out, int out_size, void* d_ws, size_t ws_size,
                              hipStream_t stream) {
  const float* stu = (const float*)d_in[0];
  const float* tea = (const float*)d_in[1];
  const float* cbk = (const float*)d_in[2];
  const int*   cod = (const int*)d_in[3];   // integer input per harness convention
  const float* org = (const float*)d_in[4];
  float* out = (float*)d_out;

  // workspace layout (~33.1 MB of floats)
  float* zf       = (float*)d_ws;                  // 12000*512
  float* cbT2     = zf + (size_t)NN * CC;          // 512*4096 (c-pair interleaved)
  float* c2       = cbT2 + (size_t)CC * KK;        // 4096
  float* ce       = c2 + KK;                       // 12000
  int*   hidx     = (int*)(ce + NN);               // 12000
  float* partials = (float*)(hidx + NN);           // 47*8

  c2_kernel<<<KK / 8, 256, 0, stream>>>(cbk, c2);
  cbt_kernel<<<dim3(KK / 64, CC / 64), 256, 0, stream>>>(cbk, cbT2);
  zt_kernel<<<dim3((TT + 63) / 64, CC / 64, NB), 256, 0, stream>>>(stu, zf);
  gemm_kernel<<<NN / 16, 256, 0, stream>>>(zf, cbT2, c2, cod, ce, hidx);
  rowloss_kernel<<<NBLK_ROWLOSS, 256, 0, stream>>>(stu, tea, org, cbk, ce, hidx, partials);
  final_kernel<<<1, 32, 0, stream>>>(partials, out);
}